// HausdorffDistanceLoss2_57707180589516
// MI455X (gfx1250) — compile-verified
//
#include <hip/hip_runtime.h>
#include <hip/hip_bf16.h>
#include <math.h>

// Problem constants (match reference setup_inputs)
#define B_  8
#define C_  4
#define H_  256
#define W_  256
#define BC_ (B_ * C_)
#define IMG_ (H_ * W_)
#define TOT_ (B_ * C_ * H_ * W_)
#define IGNORE_IDX 255
#define BIGF 1000000.0f

typedef float v2f __attribute__((ext_vector_type(2)));
typedef float v8f __attribute__((ext_vector_type(8)));

// Async global->LDS path (CDNA5): gate on builtin availability so the file
// compiles on any toolchain; fallback is plain VGPR loads + ds stores.
#if defined(__HIP_DEVICE_COMPILE__) && defined(__gfx1250__) && \
    __has_builtin(__builtin_amdgcn_global_load_async_to_lds_b32) && \
    __has_builtin(__builtin_amdgcn_s_wait_asynccnt)
#define HD_ASYNC_LDS 1
// Builtin prototype (from hipcc diagnostic): (as1 int*, as3 int*, imm, imm)
#define AS1_INT_P __attribute__((address_space(1))) int*
#define AS3_INT_P __attribute__((address_space(3))) int*
#else
#define HD_ASYNC_LDS 0
#endif

// -------- kernel 0: zero fg-any flags --------------------------------------
__global__ void hd_zero_flags(unsigned* __restrict__ flags) {
    if (threadIdx.x < BC_) flags[threadIdx.x] = 0u;
}

// -------- kernel 1: row pass ------------------------------------------------
// One block per (b, y). Computes softmax, the 8 masks (pred fg c0..3, gt fg
// c0..3) and exact 1-D distance-to-False (left & right, with the reference's
// idx+BIG / BIG-idx sentinel semantics), then writes h^2 transposed [bc][x][y].
__global__ void hd_rowpass(const float* __restrict__ pred,
                           const int*   __restrict__ target,
                           float* __restrict__ h2_pf, float* __restrict__ h2_pb,
                           float* __restrict__ h2_gf, float* __restrict__ h2_gb,
                           unsigned* __restrict__ flags) {
    __shared__ unsigned shMask[W_];     // 8 mask bits per x
    __shared__ int dl[16][W_];          // left runs: rows 0-7 fg, 8-15 bg
    __shared__ int dr[16][W_];          // right runs
    __shared__ unsigned shAny;

    const int x  = threadIdx.x;
    const int by = blockIdx.x;
    const int b  = by / H_;
    const int y  = by % H_;
    if (x == 0) shAny = 0u;
    __syncthreads();                    // init visible before atomicOr below

    const float* pp = pred + ((size_t)b * C_ * H_ + y) * W_ + x;
    float p0 = pp[0 * IMG_], p1 = pp[1 * IMG_], p2 = pp[2 * IMG_], p3 = pp[3 * IMG_];
    int   t  = target[((size_t)b * H_ + y) * W_ + x];
    float mx = fmaxf(fmaxf(p0, p1), fmaxf(p2, p3));
    float e0 = __expf(p0 - mx), e1 = __expf(p1 - mx), e2 = __expf(p2 - mx), e3 = __expf(p3 - mx);
    float inv = 1.0f / (e0 + e1 + e2 + e3);
    float pr[4] = {e0 * inv, e1 * inv, e2 * inv, e3 * inv};
    bool valid = (t != IGNORE_IDX);

    unsigned mk = 0u;
#pragma unroll
    for (int c = 0; c < C_; ++c) {
        float pm = valid ? pr[c] : 0.0f;
        if (pm > 0.5f)      mk |= (1u << c);        // pred fg
        if (valid && t == c) mk |= (1u << (4 + c)); // gt fg (onehot*valid)
    }
    shMask[x] = mk;
    if (mk) atomicOr(&shAny, mk);
    __syncthreads();

    // 16 scanner threads: mask mm = tid&7, dir = tid>>3 (0=L,1=R).
    // Sentinel init reproduces reference: no-False-left  -> dist = 1e6 + x,
    //                                     no-False-right -> dist = 1e6 - x.
    if (threadIdx.x < 16) {
        int mm  = threadIdx.x & 7;
        int dir = threadIdx.x >> 3;
        int cf  = dir ? (1000000 - W_) : (1000000 - 1);
        int cb  = cf;
        for (int i = 0; i < W_; ++i) {
            int xx  = dir ? (W_ - 1 - i) : i;
            int bit = (shMask[xx] >> mm) & 1;
            cf = bit ? (cf + 1) : 0;   // fg mask: dist to nearest False
            cb = bit ? 0 : (cb + 1);   // bg mask = ~fg
            if (dir == 0) { dl[mm][xx] = cf; dl[8 + mm][xx] = cb; }
            else          { dr[mm][xx] = cf; dr[8 + mm][xx] = cb; }
        }
    }
    // fg.any() flags per (b,c): bit0 = pred-any, bit1 = gt-any
    if (x < C_) {
        unsigned a = shAny;
        unsigned f = ((a >> x) & 1u) | (((a >> (4 + x)) & 1u) << 1);
        if (f) atomicOr(&flags[b * C_ + x], f);
    }
    __syncthreads();

#pragma unroll
    for (int c = 0; c < C_; ++c) {
        size_t o = ((size_t)(b * C_ + c) * W_ + x) * H_ + y;   // transposed
        float hpf = fminf((float)min(dl[c][x],      dr[c][x]),      BIGF);
        float hpb = fminf((float)min(dl[8 + c][x],  dr[8 + c][x]),  BIGF);
        float hgf = fminf((float)min(dl[4 + c][x],  dr[4 + c][x]),  BIGF);
        float hgb = fminf((float)min(dl[12 + c][x], dr[12 + c][x]), BIGF);
        h2_pf[o] = hpf * hpf;
        h2_pb[o] = hpb * hpb;
        h2_gf[o] = hgf * hgf;
        h2_gb[o] = hgb * hgb;
    }
}

// -------- kernel 2: column envelope ----------------------------------------
// One block per (bc, x). d2[i] = min_k (i-k)^2 + h2[k]; uniform LDS reads
// (hardware broadcast). Columns staged in LDS via CDNA5 async global->LDS
// loads (ASYNCcnt) when available. Writes dtm in place over h2_pf / h2_gf.
__global__ void hd_colpass(float* __restrict__ h2_pf, const float* __restrict__ h2_pb,
                           float* __restrict__ h2_gf, const float* __restrict__ h2_gb,
                           const unsigned* __restrict__ flags) {
    __shared__ float s_pf[H_], s_pb[H_], s_gf[H_], s_gb[H_];
    const int y   = threadIdx.x;
    const int bcx = blockIdx.x;
    const int bc  = bcx / W_;
    const int x   = bcx % W_;
    const size_t base = ((size_t)bc * W_ + x) * H_;

#if HD_ASYNC_LDS
    // Per-lane async copy: global[base+y] -> LDS s_*[y], no VGPR round-trip.
    __builtin_amdgcn_global_load_async_to_lds_b32(
        (AS1_INT_P)(h2_pf + base + y), (AS3_INT_P)&s_pf[y], 0, 0);
    __builtin_amdgcn_global_load_async_to_lds_b32(
        (AS1_INT_P)(h2_pb + base + y), (AS3_INT_P)&s_pb[y], 0, 0);
    __builtin_amdgcn_global_load_async_to_lds_b32(
        (AS1_INT_P)(h2_gf + base + y), (AS3_INT_P)&s_gf[y], 0, 0);
    __builtin_amdgcn_global_load_async_to_lds_b32(
        (AS1_INT_P)(h2_gb + base + y), (AS3_INT_P)&s_gb[y], 0, 0);
    __builtin_amdgcn_s_wait_asynccnt(0);
#else
    s_pf[y] = h2_pf[base + y];
    s_pb[y] = h2_pb[base + y];
    s_gf[y] = h2_gf[base + y];
    s_gb[y] = h2_gb[base + y];
#endif
    const unsigned fl = flags[bc];
    __syncthreads();

    const float fy = (float)y;
    float mpf = 3.0e12f, mpb = 3.0e12f, mgf = 3.0e12f, mgb = 3.0e12f;
#pragma unroll 4
    for (int k = 0; k < H_; ++k) {
        float d  = fy - (float)k;
        float d2 = d * d;
        mpf = fminf(mpf, d2 + s_pf[k]);
        mpb = fminf(mpb, d2 + s_pb[k]);
        mgf = fminf(mgf, d2 + s_gf[k]);
        mgb = fminf(mgb, d2 + s_gb[k]);
    }
    float pd = sqrtf(mpf) + sqrtf(mpb);
    float gd = floorf(sqrtf(mgf) + sqrtf(mgb));
    if (!(fl & 1u)) pd = 0.0f;     // where(fg.any(), d, 0) for pred image
    if (!(fl & 2u)) gd = 0.0f;     // for gt image
    h2_pf[base + y] = pd;          // reuse buffer: pred_dtm
    h2_gf[base + y] = gd;          // reuse buffer: gt_dtm (floored)
}

// -------- kernel 3: loss partials with WMMA wave reduction -----------------
// One thread per (b,y,x) pixel; loops c=0..3. Wave-level reduction uses
// V_WMMA_F32_16X16X4_F32 with B = ones: D[m][*] = s_m + s_{m+16}.
__global__ void hd_loss(const float* __restrict__ pred,
                        const int*   __restrict__ target,
                        const float* __restrict__ pdtm,   // transposed [bc][x][y]
                        const float* __restrict__ gdtm,
                        float* __restrict__ partials) {
    __shared__ float blockAcc;
    if (threadIdx.x == 0) blockAcc = 0.0f;
    __syncthreads();

    const int tid = blockIdx.x * blockDim.x + threadIdx.x;   // 0 .. B*H*W-1
    const int x = tid % W_;
    const int y = (tid / W_) % H_;
    const int b = tid / (H_ * W_);

    const float* pp = pred + ((size_t)b * C_ * H_ + y) * W_ + x;
    float p0 = pp[0 * IMG_], p1 = pp[1 * IMG_], p2 = pp[2 * IMG_], p3 = pp[3 * IMG_];
    int   t  = target[tid];
    float mx = fmaxf(fmaxf(p0, p1), fmaxf(p2, p3));
    float e[4] = {__expf(p0 - mx), __expf(p1 - mx), __expf(p2 - mx), __expf(p3 - mx)};
    float inv = 1.0f / (e[0] + e[1] + e[2] + e[3]);
    int tc = t < 0 ? 0 : (t > C_ - 1 ? C_ - 1 : t);          // clip for one-hot

    float acc = 0.0f;
#pragma unroll
    for (int c = 0; c < C_; ++c) {
        float prob = e[c] * inv;
        float err  = prob - (tc == c ? 1.0f : 0.0f);
        err *= err;
        size_t o = ((size_t)(b * C_ + c) * W_ + x) * H_ + y;
        float pdv = pdtm[o];
        float gdv = gdtm[o];
        acc += err * (pdv * pdv + gdv * gdv);
    }

    // ---- wave32 reduction via WMMA ones-trick (EXEC is all-1s here) ----
    v2f a;  a.x = acc;  a.y = 0.0f;      // A 16x4: A[m][0]=s_m (lo lanes), A[m][2]=s_{16+m}
    v2f bo; bo.x = 1.0f; bo.y = 1.0f;    // B 4x16 = ones
    v8f cz = {};
    v8f dm = __builtin_amdgcn_wmma_f32_16x16x4_f32(false, a, false, bo,
                                                   (short)0, cz, false, false);
    // lane<16 holds D[M=0..7][*], lane>=16 holds D[M=8..15][*]; D[m][*] = s_m+s_{m+16}
    float halfsum = dm[0] + dm[1] + dm[2] + dm[3] + dm[4] + dm[5] + dm[6] + dm[7];
    unsigned lane = threadIdx.x & 31u;
    if (lane == 0u || lane == 16u) atomicAdd(&blockAcc, halfsum);
    __syncthreads();
    if (threadIdx.x == 0) partials[blockIdx.x] = blockAcc;
}

// -------- kernel 4: deterministic final reduce -----------------------------
__global__ void hd_final(const float* __restrict__ partials, float* __restrict__ out, int n) {
    __shared__ float sh[256];
    float s = 0.0f;
    for (int i = threadIdx.x; i < n; i += 256) s += partials[i];
    sh[threadIdx.x] = s;
    __syncthreads();
    for (int st = 128; st > 0; st >>= 1) {
        if (threadIdx.x < st) sh[threadIdx.x] += sh[threadIdx.x + st];
        __syncthreads();
    }
    if (threadIdx.x == 0)
        out[0] = sh[0] * (1.0f / (float)((size_t)B_ * H_ * W_ * C_));  // 2^-21
}

extern "C" void kernel_launch(void* const* d_in, const int* in_sizes, int n_in,
                              void* d_out, int out_size, void* d_ws, size_t ws_size,
                              hipStream_t stream) {
    const float* pred   = (const float*)d_in[0];
    const int*   target = (const int*)d_in[1];
    float* out = (float*)d_out;

    // workspace layout: 4 x TOT_ floats for h2 (pf/gf reused as dtm outputs),
    // then block partials, then fg-any flags. ~32 MB total.
    float* h2_pf = (float*)d_ws;
    float* h2_pb = h2_pf + TOT_;
    float* h2_gf = h2_pb + TOT_;
    float* h2_gb = h2_gf + TOT_;
    const int NBLK = (B_ * H_ * W_) / 256;          // 2048
    float*    partials = h2_gb + TOT_;
    unsigned* flags    = (unsigned*)(partials + NBLK);

    hd_zero_flags<<<1, 64, 0, stream>>>(flags);
    hd_rowpass<<<B_ * H_, W_, 0, stream>>>(pred, target, h2_pf, h2_pb, h2_gf, h2_gb, flags);
    hd_colpass<<<BC_ * W_, H_, 0, stream>>>(h2_pf, h2_pb, h2_gf, h2_gb, flags);
    hd_loss<<<NBLK, 256, 0, stream>>>(pred, target, h2_pf, h2_gf, partials);
    hd_final<<<1, 256, 0, stream>>>(partials, out, NBLK);
}